// ADMEPredictor_84524956385586
// MI455X (gfx1250) — compile-verified
//
#include <hip/hip_runtime.h>
#include <hip/hip_bf16.h>

typedef __attribute__((ext_vector_type(2))) float v2f;
typedef __attribute__((ext_vector_type(8))) float v8f;

// ---------------------------------------------------------------------------
// Utility kernels
// ---------------------------------------------------------------------------
__global__ void fill_f(float* __restrict__ p, float v, int n) {
    int i = blockIdx.x * blockDim.x + threadIdx.x;
    if (i < n) p[i] = v;
}

__global__ void deg_edges(const int* __restrict__ dst, float* __restrict__ deg, int nE) {
    int e = blockIdx.x * blockDim.x + threadIdx.x;
    if (e < nE) atomicAdd(&deg[dst[e]], 1.0f);
}

__global__ void rsqrt_inplace(float* __restrict__ d, int n) {
    int i = blockIdx.x * blockDim.x + threadIdx.x;
    if (i < n) d[i] = rsqrtf(d[i]);   // deg >= 1 always (self loop)
}

// ---------------------------------------------------------------------------
// fp32 WMMA GEMM:  out[N, DOUT] = H[N, DIN] @ W[DIN, DOUT]
// One wave computes a 16x16 output tile; waves in a block share a 16-row strip.
// blockDim.x = (DOUT/16) * 32
// ---------------------------------------------------------------------------
template <int DIN, int DOUT>
__global__ void gemm_wmma_f32(const float* __restrict__ H,
                              const float* __restrict__ W,
                              float* __restrict__ out, int nrows) {
    constexpr int APAD = DIN + 4;              // kill stride-DIN bank conflicts
    __shared__ float Atile[16 * APAD];

    const int tid   = threadIdx.x;
    const int row0  = blockIdx.x * 16;

    // Cooperatively stage the 16 x DIN A-tile (coalesced global reads).
    for (int i = tid; i < 16 * DIN; i += blockDim.x) {
        int r = i / DIN, c = i % DIN;
        int row = row0 + r;
        if (row >= nrows) row = nrows - 1;     // clamp, store is guarded later
        Atile[r * APAD + c] = H[(size_t)row * DIN + c];
    }
    __syncthreads();

    const int wave = tid >> 5;                 // column tile index
    const int lane = tid & 31;
    const int n0   = wave * 16;
    const int hi   = lane >> 4;                // 0 or 1
    const int lo   = lane & 15;

    v8f c = {};
    #pragma unroll
    for (int k0 = 0; k0 < DIN; k0 += 4) {
        const int k = k0 + hi * 2;
        v2f a, b;
        // A 16x4 frag: VGPR r -> K = k0 + r + 2*hi, M = lo
        a.x = Atile[lo * APAD + k + 0];
        a.y = Atile[lo * APAD + k + 1];
        // B 4x16 frag: VGPR r -> K = k0 + r + 2*hi, N = lo  (W hot in WGP$/L2)
        b.x = W[(size_t)(k + 0) * DOUT + n0 + lo];
        b.y = W[(size_t)(k + 1) * DOUT + n0 + lo];
        c = __builtin_amdgcn_wmma_f32_16x16x4_f32(
                /*neg_a=*/false, a, /*neg_b=*/false, b,
                /*c_mod=*/(short)0, c, /*reuse_a=*/false, /*reuse_b=*/false);
    }

    // C/D: VGPR r -> M = r + 8*hi, N = lo
    #pragma unroll
    for (int r = 0; r < 8; ++r) {
        int row = row0 + r + hi * 8;
        if (row < nrows) out[(size_t)row * DOUT + n0 + lo] = c[r];
    }
}

// ---------------------------------------------------------------------------
// Aggregation: agg = dinv^2 * hw (self loop), += norm * hw[src] over edges,
// then h = relu(agg + bias)
// ---------------------------------------------------------------------------
template <int D>
__global__ void self_agg(const float* __restrict__ hw, const float* __restrict__ dinv,
                         float* __restrict__ agg, int n) {
    int t = blockIdx.x * blockDim.x + threadIdx.x;
    if (t < n * D) {
        int i = t / D;
        float dv = dinv[i];
        agg[t] = hw[t] * dv * dv;
    }
}

template <int D>
__global__ void edge_agg(const float* __restrict__ hw,
                         const int* __restrict__ src, const int* __restrict__ dst,
                         const float* __restrict__ dinv,
                         float* __restrict__ agg, int nE) {
    int t = blockIdx.x * blockDim.x + threadIdx.x;
    int e = t / D, f = t % D;
    if (e < nE) {
        int s = src[e], d = dst[e];
        float nrm = dinv[s] * dinv[d];
        atomicAdd(&agg[(size_t)d * D + f], nrm * hw[(size_t)s * D + f]);
    }
}

template <int D>
__global__ void bias_relu(float* __restrict__ h, const float* __restrict__ b, int n) {
    int t = blockIdx.x * blockDim.x + threadIdx.x;
    if (t < n * D) {
        float v = h[t] + b[t % D];
        h[t] = v > 0.0f ? v : 0.0f;
    }
}

// ---------------------------------------------------------------------------
// Global mean pool (D = 64)
// ---------------------------------------------------------------------------
__global__ void pool_scatter(const float* __restrict__ h, const int* __restrict__ batch,
                             float* __restrict__ pooled, float* __restrict__ cnt, int n) {
    int t = blockIdx.x * blockDim.x + threadIdx.x;
    int i = t >> 6, f = t & 63;
    if (i < n) {
        int g = batch[i];
        atomicAdd(&pooled[(size_t)g * 64 + f], h[(size_t)i * 64 + f]);
        if (f == 0) atomicAdd(&cnt[g], 1.0f);
    }
}

__global__ void pool_div(float* __restrict__ pooled, const float* __restrict__ cnt, int G) {
    int t = blockIdx.x * blockDim.x + threadIdx.x;
    if (t < G * 64) {
        float c = cnt[t >> 6];
        pooled[t] /= (c > 1.0f ? c : 1.0f);
    }
}

// ---------------------------------------------------------------------------
// 5 task heads: one wave per (t, g); lane k owns hidden unit k (K = 32 lanes)
// ---------------------------------------------------------------------------
__global__ void heads(const float* __restrict__ pooled,
                      const float* __restrict__ Wa, const float* __restrict__ ba,
                      const float* __restrict__ Wb, const float* __restrict__ bb,
                      float* __restrict__ out, int G) {
    int w = (blockIdx.x * blockDim.x + threadIdx.x) >> 5;
    int lane = threadIdx.x & 31;
    if (w >= 5 * G) return;
    int t = w / G, g = w % G;

    float hid = ba[t * 32 + lane];
    const float* p  = pooled + (size_t)g * 64;
    const float* wa = Wa + (size_t)t * 64 * 32 + lane;
    #pragma unroll
    for (int d = 0; d < 64; ++d) hid = fmaf(p[d], wa[d * 32], hid);
    hid = hid > 0.0f ? hid : 0.0f;

    float v = hid * Wb[t * 32 + lane];
    #pragma unroll
    for (int off = 16; off > 0; off >>= 1) v += __shfl_xor(v, off);
    if (lane == 0) out[(size_t)t * G + g] = v + bb[t];
}

// ---------------------------------------------------------------------------
// Launcher
// ---------------------------------------------------------------------------
extern "C" void kernel_launch(void* const* d_in, const int* in_sizes, int n_in,
                              void* d_out, int out_size, void* d_ws, size_t ws_size,
                              hipStream_t stream) {
    const float* x     = (const float*)d_in[0];
    const int*   ei    = (const int*)d_in[1];
    const int*   batch = (const int*)d_in[2];
    const float* W1 = (const float*)d_in[3];
    const float* b1 = (const float*)d_in[4];
    const float* W2 = (const float*)d_in[5];
    const float* b2 = (const float*)d_in[6];
    const float* W3 = (const float*)d_in[7];
    const float* b3 = (const float*)d_in[8];
    const float* Wa = (const float*)d_in[9];
    const float* ba = (const float*)d_in[10];
    const float* Wb = (const float*)d_in[11];
    const float* bb = (const float*)d_in[12];
    float* out = (float*)d_out;

    const int N = in_sizes[0] / 64;     // 400000
    const int E = in_sizes[1] / 2;      // 1600000
    const int G = out_size / 5;         // 16000
    const int* src = ei;
    const int* dst = ei + E;

    // Workspace layout (floats)
    float* ws = (float*)d_ws;
    size_t off = 0;
    float* dinv = ws + off; off += (size_t)N;       off = (off + 255) & ~(size_t)255;
    float* bufA = ws + off; off += (size_t)N * 128; // hw
    float* bufB = ws + off; off += (size_t)N * 128; // agg / h
    float* pooled = ws + off; off += (size_t)G * 64;
    float* cnt    = ws + off; off += (size_t)G;
    (void)ws_size; (void)n_in;

    const int B = 256;
    auto cdiv = [](long long a, long long b) { return (int)((a + b - 1) / b); };

    // ---- normalization: dinv = rsqrt(deg + self loop) ----
    fill_f<<<cdiv(N, B), B, 0, stream>>>(dinv, 1.0f, N);
    deg_edges<<<cdiv(E, B), B, 0, stream>>>(dst, dinv, E);
    rsqrt_inplace<<<cdiv(N, B), B, 0, stream>>>(dinv, N);

    const int gemmGrid = cdiv(N, 16);

    // ---- layer 1: [N,64] @ [64,128] ----
    gemm_wmma_f32<64, 128><<<gemmGrid, (128 / 16) * 32, 0, stream>>>(x, W1, bufA, N);
    self_agg<128><<<cdiv((long long)N * 128, B), B, 0, stream>>>(bufA, dinv, bufB, N);
    edge_agg<128><<<cdiv((long long)E * 128, B), B, 0, stream>>>(bufA, src, dst, dinv, bufB, E);
    bias_relu<128><<<cdiv((long long)N * 128, B), B, 0, stream>>>(bufB, b1, N);

    // ---- layer 2: [N,128] @ [128,128] ----
    gemm_wmma_f32<128, 128><<<gemmGrid, (128 / 16) * 32, 0, stream>>>(bufB, W2, bufA, N);
    self_agg<128><<<cdiv((long long)N * 128, B), B, 0, stream>>>(bufA, dinv, bufB, N);
    edge_agg<128><<<cdiv((long long)E * 128, B), B, 0, stream>>>(bufA, src, dst, dinv, bufB, E);
    bias_relu<128><<<cdiv((long long)N * 128, B), B, 0, stream>>>(bufB, b2, N);

    // ---- layer 3: [N,128] @ [128,64] ----
    gemm_wmma_f32<128, 64><<<gemmGrid, (64 / 16) * 32, 0, stream>>>(bufB, W3, bufA, N);
    self_agg<64><<<cdiv((long long)N * 64, B), B, 0, stream>>>(bufA, dinv, bufB, N);
    edge_agg<64><<<cdiv((long long)E * 64, B), B, 0, stream>>>(bufA, src, dst, dinv, bufB, E);
    bias_relu<64><<<cdiv((long long)N * 64, B), B, 0, stream>>>(bufB, b3, N);

    // ---- global mean pool ----
    fill_f<<<cdiv((long long)G * 64 + G, B), B, 0, stream>>>(pooled, 0.0f, G * 64 + G); // pooled+cnt contiguous
    pool_scatter<<<cdiv((long long)N * 64, B), B, 0, stream>>>(bufB, batch, pooled, cnt, N);
    pool_div<<<cdiv((long long)G * 64, B), B, 0, stream>>>(pooled, cnt, G);

    // ---- 5 task heads ----
    heads<<<cdiv((long long)5 * G * 32, B), B, 0, stream>>>(pooled, Wa, ba, Wb, bb, out, G);
}